// RotRandomizer_14216341749812
// MI455X (gfx1250) — compile-verified
//
#include <hip/hip_runtime.h>
#include <math.h>

// Problem constants (from reference setup_inputs)
#define B_   32
#define T_   8
#define C_   3
#define H_   224
#define W_   224
#define K_   64
#define DA   10
#define HW_  (H_*W_)            // 50176
#define TC_  (T_*C_)            // 24

// Flat output layout: obs_out | pos_out | quat_out | act_out
#define OBS_N    38535168       // 32*8*3*224*224
#define POS_OFF  38535168       // + B*T*3  = 768
#define QUAT_OFF 38535936       // + B*T*4  = 1024
#define ACT_OFF  38536960       // + B*T*10 = 2560

// 4-byte-aligned vector types so odd-column b64 gathers stay single loads
typedef float vfloat2 __attribute__((ext_vector_type(2), aligned(4)));
typedef float vfloat4 __attribute__((ext_vector_type(4), aligned(16)));

// ------------------------------------------------------------------
// Kernel A: angle candidates, validity reduction, selection, and all
// small outputs (pos/quat/act). One block, 256 threads (8 waves).
// Publishes c_sel[32], s_sel[32], any_valid to workspace.
// ------------------------------------------------------------------
__global__ __launch_bounds__(256) void rot_select_kernel(
    const float* __restrict__ pos,      // (B,T,3)
    const float* __restrict__ quat,     // (B,T,4)
    const float* __restrict__ naction,  // (B,T,10)
    const float* __restrict__ angles_u, // (K,B)
    const float* __restrict__ zero_u,   // (K,B)
    float* __restrict__ out,
    float* __restrict__ ws)
{
    __shared__ float cA[K_*B_];
    __shared__ float sA[K_*B_];
    __shared__ int   okA[K_];
    __shared__ int   bad;
    __shared__ int   selIdx;
    __shared__ int   anyV;

    const int tid = threadIdx.x;

    const float TWO_PI = 6.28318530717958647692f;
    const float PI_    = 3.14159265358979323846f;

    // cos/sin for all (k,b) candidates
    for (int i = tid; i < K_*B_; i += 256) {
        float a = angles_u[i] * TWO_PI - PI_;
        if (zero_u[i] < 0.015625f) a = 0.0f;   // 1/64
        float sv, cv;
        sincosf(a, &sv, &cv);
        cA[i] = cv;
        sA[i] = sv;
    }
    __syncthreads();

    // one thread per (b,t)
    const int b  = tid >> 3;
    const int tt = tid & 7;
    const int bt = b * T_ + tt;          // == tid

    const float px = pos[bt*3+0];
    const float py = pos[bt*3+1];
    const float pz = pos[bt*3+2];
    const float n0 = naction[bt*DA+0];
    const float n1 = naction[bt*DA+1];

    // per-k validity: AND-reduce across all (b,t) via shared atomicOr
    for (int k = 0; k < K_; ++k) {
        if (tid == 0) bad = 0;
        __syncthreads();
        const float c = cA[k*B_ + b];
        const float s = sA[k*B_ + b];
        const float rx = c*px - s*py;
        const float ry = s*px + c*py;
        const float a0 = c*n0 - s*n1;
        const float a1 = s*n0 + c*n1;
        // pos: closed interval [-1,1]; naction[:2]: open interval (-1,1)
        const bool viol = (rx < -1.0f) || (rx > 1.0f) ||
                          (ry < -1.0f) || (ry > 1.0f) ||
                          (pz < -1.0f) || (pz > 1.0f) ||
                          (a0 <= -1.0f) || (a0 >= 1.0f) ||
                          (a1 <= -1.0f) || (a1 >= 1.0f);
        if (viol) atomicOr(&bad, 1);
        __syncthreads();
        if (tid == 0) okA[k] = (bad == 0);
    }
    __syncthreads();

    if (tid == 0) {
        int idx = 0, any = 0;
        for (int k = 0; k < K_; ++k) {
            if (okA[k]) { idx = k; any = 1; break; }   // argmax of bool = first True
        }
        selIdx = idx;
        anyV   = any;
        ((int*)ws)[64] = any;
    }
    __syncthreads();

    const int idx = selIdx;
    const int any = anyV;

    if (tid < B_) {
        ws[tid]      = cA[idx*B_ + tid];
        ws[32 + tid] = sA[idx*B_ + tid];
    }

    const float c = cA[idx*B_ + b];
    const float s = sA[idx*B_ + b];

    float on[10];
    #pragma unroll
    for (int j = 0; j < 10; ++j) on[j] = naction[bt*DA + j];
    const float qx = quat[bt*4+0], qy = quat[bt*4+1];
    const float qz = quat[bt*4+2], qw = quat[bt*4+3];

    if (any) {
        // ---- pos_out
        out[POS_OFF + bt*3+0] = c*px - s*py;
        out[POS_OFF + bt*3+1] = s*px + c*py;
        out[POS_OFF + bt*3+2] = pz;

        // ---- act_out (rn layout from reference)
        float rn[10];
        rn[0] = c*on[0] - s*on[1];
        rn[1] = s*on[0] + c*on[1];
        rn[2] = on[2];
        rn[3] = c*on[3] - s*on[6];
        rn[4] = c*on[4] - s*on[7];
        rn[5] = c*on[5] - s*on[8];
        rn[6] = s*on[3] + c*on[6];
        rn[7] = s*on[4] + c*on[7];
        rn[8] = s*on[5] + c*on[8];
        rn[9] = on[9];
        #pragma unroll
        for (int j = 0; j < 10; ++j) out[ACT_OFF + bt*DA + j] = rn[j];

        // ---- quat_out: q(xyzw) -> wxyz -> matrix -> R3*M -> quaternion -> xyzw
        const float r = qw, i = qx, j2 = qy, kq = qz;
        const float nrm = r*r + i*i + j2*j2 + kq*kq;
        const float ts  = 2.0f / nrm;
        const float m00 = 1.0f - ts*(j2*j2 + kq*kq);
        const float m01 = ts*(i*j2 - kq*r);
        const float m02 = ts*(i*kq + j2*r);
        const float m10 = ts*(i*j2 + kq*r);
        const float m11 = 1.0f - ts*(i*i + kq*kq);
        const float m12 = ts*(j2*kq - i*r);
        const float m20 = ts*(i*kq - j2*r);
        const float m21 = ts*(j2*kq + i*r);
        const float m22 = 1.0f - ts*(i*i + j2*j2);

        const float r00 = c*m00 - s*m10, r01 = c*m01 - s*m11, r02 = c*m02 - s*m12;
        const float r10 = s*m00 + c*m10, r11 = s*m01 + c*m11, r12 = s*m02 + c*m12;
        const float r20 = m20, r21 = m21, r22 = m22;

        const float qa0 = sqrtf(fmaxf(1.0f + r00 + r11 + r22, 0.0f));
        const float qa1 = sqrtf(fmaxf(1.0f + r00 - r11 - r22, 0.0f));
        const float qa2 = sqrtf(fmaxf(1.0f - r00 + r11 - r22, 0.0f));
        const float qa3 = sqrtf(fmaxf(1.0f - r00 - r11 + r22, 0.0f));

        const float rows[4][4] = {
            { qa0*qa0,   r21 - r12, r02 - r20, r10 - r01 },
            { r21 - r12, qa1*qa1,   r10 + r01, r02 + r20 },
            { r02 - r20, r10 + r01, qa2*qa2,   r12 + r21 },
            { r10 - r01, r20 + r02, r21 + r12, qa3*qa3   }
        };
        int   im   = 0;
        float best = qa0;
        if (qa1 > best) { best = qa1; im = 1; }
        if (qa2 > best) { best = qa2; im = 2; }
        if (qa3 > best) { best = qa3; im = 3; }
        const float den = 2.0f * fmaxf(best, 0.1f);
        // (r,i,j,k) reindexed [1,2,3,0] -> (i,j,k,r)
        out[QUAT_OFF + bt*4+0] = rows[im][1] / den;
        out[QUAT_OFF + bt*4+1] = rows[im][2] / den;
        out[QUAT_OFF + bt*4+2] = rows[im][3] / den;
        out[QUAT_OFF + bt*4+3] = rows[im][0] / den;
    } else {
        out[POS_OFF + bt*3+0] = px;
        out[POS_OFF + bt*3+1] = py;
        out[POS_OFF + bt*3+2] = pz;
        out[QUAT_OFF + bt*4+0] = qx;
        out[QUAT_OFF + bt*4+1] = qy;
        out[QUAT_OFF + bt*4+2] = qz;
        out[QUAT_OFF + bt*4+3] = qw;
        #pragma unroll
        for (int j = 0; j < 10; ++j) out[ACT_OFF + bt*DA + j] = on[j];
    }
}

// ------------------------------------------------------------------
// Kernel B: rotated bilinear grid-sample (zero padding, align_corners).
// Per pixel: two b64 gathers (x-adjacent corners are contiguous) and
// bilinear blend; 4 pixels/thread, one NT b128 store. Bandwidth-bound:
// ~308 MB @ 23.3 TB/s => ~13 us floor; obs (154 MB) is kept resident
// in the 192 MB L2 by streaming the write-once output with TH=NT.
// ------------------------------------------------------------------
__device__ __forceinline__ void fetch_pair(const float* __restrict__ base,
                                           int ix0, int iy,
                                           float& vA, float& vB)
{
    // values at (ix0, iy) and (ix0+1, iy), zero outside [0,W)x[0,H)
    const bool iyok = ((unsigned)iy < (unsigned)H_);
    const int  iyc  = min(max(iy, 0), H_-1);
    const int  ixl  = min(max(ix0, 0), W_-2);          // pair load stays in-row
    const vfloat2 p = *(const vfloat2*)(base + iyc*W_ + ixl);
    const bool aok = iyok && ((unsigned)ix0       < (unsigned)W_);
    const bool bok = iyok && ((unsigned)(ix0 + 1) < (unsigned)W_);
    // if ix0 valid, ix0 ∈ {ixl, ixl+1}; if ix0+1 valid, ix0+1 ∈ {ixl, ixl+1}
    const float va = (ix0 == ixl)     ? p.x : p.y;
    const float vb = (ix0 + 1 == ixl) ? p.x : p.y;
    vA = aok ? va : 0.0f;
    vB = bok ? vb : 0.0f;
}

__global__ __launch_bounds__(256) void grid_sample_kernel(
    const float* __restrict__ obs,   // (B, T*C, H, W)
    float* __restrict__ out,
    const float* __restrict__ ws)
{
    const int tid  = blockIdx.x * blockDim.x + threadIdx.x;
    const int gid4 = tid * 4;                 // first of 4 consecutive pixels
    // decode (bc, h, w0); groups of 4 never cross a row (224 % 4 == 0)
    const int bc = gid4 / HW_;                // b*24 + tc
    const int p  = gid4 - bc * HW_;
    const int h  = p / W_;
    const int w0 = p - h * W_;
    const int b  = bc / TC_;

    const float* __restrict__ base = obs + (size_t)bc * HW_;
    const int any = ((const int*)ws)[64];

    vfloat4 res;
    if (!any) {
        // passthrough: obs_out = obs (one-shot stream, keep out of L2)
        res = __builtin_nontemporal_load((const vfloat4*)(base + p));
    } else {
        const float c = ws[b];
        const float s = ws[32 + b];
        const float inv = 2.0f / (float)(W_ - 1);    // == 2/(H-1), square image
        const float Y  = -1.0f + (float)h  * inv;
        const float X0 = -1.0f + (float)w0 * inv;
        float gx = c*X0 - s*Y;
        float gy = s*X0 + c*Y;
        const float dgx = c * inv;                   // x advances by c px/pixel
        const float dgy = s * inv;                   // y advances by s px/pixel

        // CDNA5 global_prefetch_b8: lead the gather stream (sample region is
        // spatially near the identity location for small rotations)
        __builtin_prefetch(base + p, 0, 3);

        float o[4];
        #pragma unroll
        for (int i = 0; i < 4; ++i) {
            const float x = (gx + 1.0f) * (0.5f * (float)(W_ - 1));
            const float y = (gy + 1.0f) * (0.5f * (float)(H_ - 1));
            const float x0f = floorf(x);
            const float y0f = floorf(y);
            const float wx = x - x0f;
            const float wy = y - y0f;
            const int ix0 = (int)x0f;
            const int iy0 = (int)y0f;
            float v00, v01, v10, v11;
            fetch_pair(base, ix0, iy0,     v00, v01);   // one b64
            fetch_pair(base, ix0, iy0 + 1, v10, v11);   // one b64
            o[i] = (v00 * (1.0f - wx) + v01 * wx) * (1.0f - wy)
                 + (v10 * (1.0f - wx) + v11 * wx) * wy;
            gx += dgx;
            gy += dgy;
        }
        res.x = o[0]; res.y = o[1]; res.z = o[2]; res.w = o[3];
    }
    // TH=NT store: output is written once, never re-read -> don't evict obs
    __builtin_nontemporal_store(res, (vfloat4*)(out + gid4));
}

// ------------------------------------------------------------------
extern "C" void kernel_launch(void* const* d_in, const int* in_sizes, int n_in,
                              void* d_out, int out_size, void* d_ws, size_t ws_size,
                              hipStream_t stream)
{
    (void)in_sizes; (void)n_in; (void)out_size; (void)ws_size;
    const float* obs      = (const float*)d_in[0];
    const float* pos      = (const float*)d_in[1];
    const float* quat     = (const float*)d_in[2];
    const float* naction  = (const float*)d_in[3];
    const float* angles_u = (const float*)d_in[4];
    const float* zero_u   = (const float*)d_in[5];
    float* out = (float*)d_out;
    float* ws  = (float*)d_ws;

    // A: selection + small outputs + publish (c_sel, s_sel, any) to ws
    rot_select_kernel<<<1, 256, 0, stream>>>(pos, quat, naction,
                                             angles_u, zero_u, out, ws);
    // B: 38,535,168 pixels / (256 threads * 4 px) = 37,632 blocks
    grid_sample_kernel<<<OBS_N / 1024, 256, 0, stream>>>(obs, out, ws);
}